// QOP_41790031790119
// MI455X (gfx1250) — compile-verified
//
#include <hip/hip_runtime.h>
#include <math.h>
#include <stdint.h>

// ---------------------------------------------------------------------------
// Sinkhorn-OT distance on MI455X (gfx1250).
//   * fp32 WMMA (V_WMMA_F32_16X16X4_F32) for P(M+M^T), Q(M+M^T) and the big
//     4096x4096x512 cross GEMM (43 GFLOP total -> far below matrix-core peak,
//     so full fp32 precision is free; output scalar is exp()-sensitive).
//   * GEMM staging uses GLOBAL_LOAD_ASYNC_TO_LDS_B128 (ASYNCcnt-tracked,
//     double-buffered): global->LDS with no VGPR round trip, overlapped with
//     the WMMA stream of the previous K-chunk.
//   * K (64MB) + dist (64MB) live in workspace; both fit in the 192MB L2, so
//     the 44 Sinkhorn matvec passes stream from L2, not HBM.
//   * K/dist epilogue fused into the cross GEMM (saves one 64MB round trip).
//   * Deterministic: no float atomics; reference early-exit replicated with a
//     device-side flag gating all Sinkhorn kernels.
// Workspace requirement: ~146 MB.
// ---------------------------------------------------------------------------

#define DIMK 512
#define SEQN 4096
#define TOLC 1e-5f
#define INV_SQRT_2PI 0.3989422804014327f

typedef float v2f __attribute__((ext_vector_type(2)));
typedef float v8f __attribute__((ext_vector_type(8)));

#if defined(__has_builtin) && __has_builtin(__builtin_amdgcn_s_wait_asynccnt)
#define WAIT_ASYNC(n) __builtin_amdgcn_s_wait_asynccnt(n)
#else
#define WAIT_ASYNC(n) asm volatile("s_wait_asynccnt %0" ::"i"(n) : "memory")
#endif

// Async copy of 16 bytes per lane: global -> LDS, tracked by ASYNCcnt.
// Low 32 bits of a generic pointer to __shared__ are the wave-relative LDS
// byte offset on AMDGPU (aperture bits live in the high half).
__device__ __forceinline__ void async_ld_b128(void* lds, const float* g) {
  uint32_t l = (uint32_t)(uintptr_t)lds;
  uint64_t a = (uint64_t)(uintptr_t)g;
  asm volatile("global_load_async_to_lds_b128 %0, %1, off" ::"v"(l), "v"(a)
               : "memory");
}

__device__ __forceinline__ float wave_reduce_sum(float v) {
#pragma unroll
  for (int off = 16; off > 0; off >>= 1) v += __shfl_xor(v, off, 32);
  return v;
}

// As = M + M^T  (symmetric metric; lets every GEMM use the NT form)
__global__ void k_symmetrize(const float* __restrict__ M, float* __restrict__ As) {
  int idx = blockIdx.x * blockDim.x + threadIdx.x;
  int d = idx / DIMK, e = idx % DIMK;
  As[idx] = M[idx] + M[e * DIMK + d];
}

// Fused epilogue: dist + prior + K
__device__ __forceinline__ void epi_store(int i, int j, float cross,
                                          const float* __restrict__ pMp,
                                          const float* __restrict__ qMq,
                                          float* __restrict__ Kout,
                                          float* __restrict__ Dout) {
  const float dist = pMp[i] + qMq[j] - cross;
  const float dij  = (float)(i - j);
  const float rel  = dij * (1.0f / SEQN);
  // (|rel|/denom)^2 / 2 with denom = sqrt(2)/4096  ->  (i-j)^2 / 4
  const float prior = INV_SQRT_2PI * __expf(-0.25f * dij * dij);
  const float kv = prior * __expf(-(dist - 0.1f / (rel * rel + 1.0f)) * (1.0f / 1000.0f));
  const size_t off = (size_t)i * SEQN + j;
  Kout[off] = kv;
  Dout[off] = dist;
}

// C[M,N] = A[M,K] * B[N,K]^T, fp32 WMMA 16x16x4.
// Block: 256 thr (8 waves), 64x64 tile; wave = 16 rows x 32 cols (2 WMMA tiles).
// K-chunks of 32 double-buffered through LDS via async global->LDS b128.
template <bool EPI>
__global__ __launch_bounds__(256) void k_gemm_nt(
    const float* __restrict__ A, const float* __restrict__ B,
    float* __restrict__ C, const float* __restrict__ pMp,
    const float* __restrict__ qMq, float* __restrict__ Kout,
    float* __restrict__ Dout, int Mdim, int Ndim, int Kdim) {
  __shared__ float sA[2][64 * 36];  // stride 36: 16B-aligned rows, skewed banks
  __shared__ float sB[2][64 * 36];
  const int tid  = threadIdx.x;
  const int wave = tid >> 5;
  const int lane = tid & 31;
  const int half = lane >> 4;  // 32-bit WMMA A/B: lanes16-31 hold K={2,3}
  const int l16  = lane & 15;
  const int bm = blockIdx.y * 64;
  const int bn = blockIdx.x * 64;
  const int wr = (wave & 3) * 16;
  const int wc = (wave >> 2) * 32;

  // Staging: 64x32 tile per array = 512 float4; 2 async b128 per thread each.
  const int r0 = tid >> 3;        // rows 0..31
  const int c0 = (tid & 7) * 4;   // float4 column
  const int r1 = r0 + 32;         // rows 32..63
  const float* gA0 = A + (size_t)(bm + r0) * Kdim + c0;
  const float* gA1 = A + (size_t)(bm + r1) * Kdim + c0;
  const float* gB0 = B + (size_t)(bn + r0) * Kdim + c0;
  const float* gB1 = B + (size_t)(bn + r1) * Kdim + c0;

  v8f acc0 = {};
  v8f acc1 = {};

  auto issue = [&](int buf, int k0) {
    async_ld_b128(&sA[buf][r0 * 36 + c0], gA0 + k0);
    async_ld_b128(&sA[buf][r1 * 36 + c0], gA1 + k0);
    async_ld_b128(&sB[buf][r0 * 36 + c0], gB0 + k0);
    async_ld_b128(&sB[buf][r1 * 36 + c0], gB1 + k0);
  };

  const int nc = Kdim / 32;
  issue(0, 0);
  for (int c = 0; c < nc; ++c) {
    const int buf = c & 1;
    if (c + 1 < nc) {
      issue(buf ^ 1, (c + 1) * 32);  // prefetch next chunk into other buffer
      WAIT_ASYNC(4);                 // newest 4 outstanding => chunk c landed
    } else {
      WAIT_ASYNC(0);
    }
    __syncthreads();  // chunk c visible from all waves
    const float* lA = &sA[buf][0];
    const float* lB = &sB[buf][0];
#pragma unroll
    for (int kk = 0; kk < 32; kk += 4) {
      const int ka = kk + 2 * half;  // VGPR0 -> K={0|2}, VGPR1 -> K={1|3}
      v2f a, b0, b1;
      a.x  = lA[(wr + l16) * 36 + ka];
      a.y  = lA[(wr + l16) * 36 + ka + 1];
      b0.x = lB[(wc + l16) * 36 + ka];
      b0.y = lB[(wc + l16) * 36 + ka + 1];
      b1.x = lB[(wc + 16 + l16) * 36 + ka];
      b1.y = lB[(wc + 16 + l16) * 36 + ka + 1];
      acc0 = __builtin_amdgcn_wmma_f32_16x16x4_f32(false, a, false, b0, (short)0, acc0, false, false);
      acc1 = __builtin_amdgcn_wmma_f32_16x16x4_f32(false, a, false, b1, (short)0, acc1, false, false);
    }
    __syncthreads();  // all waves done reading buf before it is overwritten
  }

#pragma unroll
  for (int v = 0; v < 8; ++v) {  // C layout: VGPR v -> row v (+8 for lanes16-31)
    const int gi  = bm + wr + v + 8 * half;
    const int gj0 = bn + wc + l16;
    const int gj1 = gj0 + 16;
    if constexpr (!EPI) {
      C[(size_t)gi * Ndim + gj0] = acc0[v];
      C[(size_t)gi * Ndim + gj1] = acc1[v];
    } else {
      epi_store(gi, gj0, acc0[v], pMp, qMq, Kout, Dout);
      epi_store(gi, gj1, acc1[v], pMp, qMq, Kout, Dout);
    }
  }
}

// out[i] = 0.5 * dot(XA[i,:], X[i,:])   (pMp / qMq)
__global__ void k_rowdot(const float* __restrict__ XA, const float* __restrict__ X,
                         float* __restrict__ out) {
  const int wave = threadIdx.x >> 5, lane = threadIdx.x & 31;
  const int row = blockIdx.x * 8 + wave;
  const float* a = XA + (size_t)row * DIMK;
  const float* b = X + (size_t)row * DIMK;
  float s = 0.f;
  for (int c = lane; c < DIMK; c += 32) s = fmaf(a[c], b[c], s);
  s = wave_reduce_sum(s);
  if (lane == 0) out[row] = 0.5f * s;
}

__global__ void k_init_phase(float* __restrict__ u, int* __restrict__ flag) {
  int i = blockIdx.x * blockDim.x + threadIdx.x;
  if (i < SEQN) u[i] = (float)(i + 1) * (1.0f / SEQN);
  if (i == 0) *flag = 0;
}

// Partial K^T @ u : grid (16, 8); fully coalesced column walk.
__global__ void k_col_part(const float* __restrict__ Km, const float* __restrict__ uvec,
                           float* __restrict__ tpart, const int* __restrict__ flag) {
  if (*flag) return;
  const int j  = blockIdx.x * blockDim.x + threadIdx.x;
  const int i0 = blockIdx.y * (SEQN / 8);
  const float* p = Km + (size_t)i0 * SEQN + j;
  float t = 0.f;
#pragma unroll 4
  for (int i = 0; i < SEQN / 8; ++i) t = fmaf(p[(size_t)i * SEQN], uvec[i0 + i], t);
  tpart[blockIdx.y * SEQN + j] = t;
}

// dst[j] = beta_j / sum(partials)   (produces w in body, v in check)
__global__ void k_col_fin(const float* __restrict__ tpart, float* __restrict__ dst,
                          const int* __restrict__ flag) {
  if (*flag) return;
  const int j = blockIdx.x * blockDim.x + threadIdx.x;
  float t = 0.f;
#pragma unroll
  for (int s = 0; s < 8; ++s) t += tpart[s * SEQN + j];
  dst[j] = ((float)(j + 1) * (1.0f / SEQN)) / t;
}

// u[i] = alpha_i / (K @ w)_i   (== 1/(ainvK @ w))
__global__ void k_row_upd(const float* __restrict__ Km, const float* __restrict__ wvec,
                          float* __restrict__ uvec, const int* __restrict__ flag) {
  if (*flag) return;
  const int wave = threadIdx.x >> 5, lane = threadIdx.x & 31;
  const int row = blockIdx.x * 8 + wave;
  const float* p = Km + (size_t)row * SEQN;
  float s = 0.f;
  for (int j = lane; j < SEQN; j += 32) s = fmaf(p[j], wvec[j], s);
  s = wave_reduce_sum(s);
  if (lane == 0) uvec[row] = ((float)(row + 1) * (1.0f / SEQN)) / s;
}

// critpart[b] = sum_j |v_j * (K^T u)_j - beta_j|  over this block's columns
__global__ void k_crit_part(const float* __restrict__ tpart, const float* __restrict__ vvec,
                            float* __restrict__ critpart, const int* __restrict__ flag) {
  if (*flag) return;
  const int tid = threadIdx.x;
  const int j = blockIdx.x * 256 + tid;
  float s = 0.f;
#pragma unroll
  for (int seg = 0; seg < 8; ++seg) s += tpart[seg * SEQN + j];
  float val = fabsf(vvec[j] * s - (float)(j + 1) * (1.0f / SEQN));
  val = wave_reduce_sum(val);
  __shared__ float red[8];
  if ((tid & 31) == 0) red[tid >> 5] = val;
  __syncthreads();
  if (tid == 0) {
    float c = 0.f;
#pragma unroll
    for (int w = 0; w < 8; ++w) c += red[w];
    critpart[blockIdx.x] = c;
  }
}

__global__ void k_set_flag(const float* __restrict__ critpart, int* __restrict__ flag) {
  if (*flag) return;
  if (threadIdx.x == 0) {
    float c = 0.f;
    for (int b = 0; b < 16; ++b) c += critpart[b];
    if (fabsf(c) < TOLC) *flag = 1;
  }
}

// rowpart[i] = u_i * sum_j K_ij * v_j * dist_ij
__global__ void k_final_row(const float* __restrict__ Km, const float* __restrict__ Dm,
                            const float* __restrict__ uvec, const float* __restrict__ vvec,
                            float* __restrict__ rowpart) {
  const int wave = threadIdx.x >> 5, lane = threadIdx.x & 31;
  const int row = blockIdx.x * 8 + wave;
  const float* pk = Km + (size_t)row * SEQN;
  const float* pd = Dm + (size_t)row * SEQN;
  float s = 0.f;
  for (int j = lane; j < SEQN; j += 32) s = fmaf(pk[j] * vvec[j], pd[j], s);
  s = wave_reduce_sum(s);
  if (lane == 0) rowpart[row] = uvec[row] * s;
}

// Single-block ordered reduction -> deterministic across graph replays.
__global__ void k_final_sum(const float* __restrict__ rowpart, float* __restrict__ out,
                            int phase) {
  const int tid = threadIdx.x;
  float s = 0.f;
  for (int i = tid; i < SEQN; i += 256) s += rowpart[i];
  s = wave_reduce_sum(s);
  __shared__ float red[8];
  if ((tid & 31) == 0) red[tid >> 5] = s;
  __syncthreads();
  if (tid == 0) {
    float tot = 0.f;
#pragma unroll
    for (int w = 0; w < 8; ++w) tot += red[w];
    out[0] = (phase == 0) ? tot : (out[0] - tot);
  }
}

extern "C" void kernel_launch(void* const* d_in, const int* in_sizes, int n_in,
                              void* d_out, int out_size, void* d_ws, size_t ws_size,
                              hipStream_t stream) {
  (void)in_sizes; (void)n_in; (void)out_size; (void)ws_size;
  const float* P = (const float*)d_in[0];
  const float* Q = (const float*)d_in[1];
  const float* R = (const float*)d_in[2];
  const float* S = (const float*)d_in[3];
  const float* M = (const float*)d_in[4];
  float* out = (float*)d_out;

  float* wsf = (float*)d_ws;
  // workspace layout (floats); total ~38.1M floats (~146 MB)
  float* As   = wsf;                       // 512*512
  float* PA   = As + DIMK * DIMK;          // 4096*512
  float* QA   = PA + SEQN * DIMK;          // 4096*512
  float* pMp  = QA + SEQN * DIMK;          // 4096
  float* qMq  = pMp + SEQN;                // 4096
  float* Km   = qMq + SEQN;                // 4096*4096
  float* Dm   = Km + (size_t)SEQN * SEQN;  // 4096*4096
  float* uvec = Dm + (size_t)SEQN * SEQN;  // 4096
  float* vvec = uvec + SEQN;               // 4096
  float* wvec = vvec + SEQN;               // 4096
  float* tpart = wvec + SEQN;              // 8*4096 (also reused as rowpart)
  float* critp = tpart + 8 * SEQN;         // 16
  int*   flag  = (int*)(critp + 16);       // 1

  const dim3 blk(256);
  const dim3 gSmall(DIMK / 64, SEQN / 64);  // (8, 64)  for P@A, Q@A
  const dim3 gBig(SEQN / 64, SEQN / 64);    // (64, 64) for cross GEMM
  const dim3 gCol(16, 8);

  // shared metric
  k_symmetrize<<<DIMK * DIMK / 256, blk, 0, stream>>>(M, As);

  auto body = [&]() {
    k_col_part<<<gCol, blk, 0, stream>>>(Km, uvec, tpart, flag);
    k_col_fin<<<16, blk, 0, stream>>>(tpart, wvec, flag);
    k_row_upd<<<SEQN / 8, blk, 0, stream>>>(Km, wvec, uvec, flag);
  };
  auto check = [&]() {
    k_col_part<<<gCol, blk, 0, stream>>>(Km, uvec, tpart, flag);
    k_col_fin<<<16, blk, 0, stream>>>(tpart, vvec, flag);           // v = beta/(K^T u)
    k_row_upd<<<SEQN / 8, blk, 0, stream>>>(Km, vvec, uvec, flag);  // u = alpha/(K v)
    k_col_part<<<gCol, blk, 0, stream>>>(Km, uvec, tpart, flag);    // s = K^T u
    k_crit_part<<<16, blk, 0, stream>>>(tpart, vvec, critp, flag);
    k_set_flag<<<1, 32, 0, stream>>>(critp, flag);
  };

  auto phase = [&](const float* Pp, const float* Qp, int ph) {
    k_gemm_nt<false><<<gSmall, blk, 0, stream>>>(Pp, As, PA, nullptr, nullptr,
                                                 nullptr, nullptr, SEQN, DIMK, DIMK);
    k_gemm_nt<false><<<gSmall, blk, 0, stream>>>(Qp, As, QA, nullptr, nullptr,
                                                 nullptr, nullptr, SEQN, DIMK, DIMK);
    k_rowdot<<<SEQN / 8, blk, 0, stream>>>(PA, Pp, pMp);
    k_rowdot<<<SEQN / 8, blk, 0, stream>>>(QA, Qp, qMq);
    // cross GEMM with fused dist/K epilogue
    k_gemm_nt<true><<<gBig, blk, 0, stream>>>(PA, Qp, nullptr, pMp, qMq, Km, Dm,
                                              SEQN, SEQN, DIMK);
    k_init_phase<<<SEQN / 256, blk, 0, stream>>>(uvec, flag);
    // reference trace: body, check(it=1), 18x body, check(it=20)
    body();
    check();
    for (int it = 0; it < 18; ++it) body();
    check();
    k_final_row<<<SEQN / 8, blk, 0, stream>>>(Km, Dm, uvec, vvec, tpart);
    k_final_sum<<<1, blk, 0, stream>>>(tpart, out, ph);
  };

  phase(P, Q, 0);  // out  = W_M(P,Q,M)
  phase(R, S, 1);  // out -= W_M(R,S,M)
}